// Net_34265249087832
// MI455X (gfx1250) — compile-verified
//
#include <hip/hip_runtime.h>

// SNN forward (snntorch Leaky, reset-by-subtract), fp32.
// Stage 1: cur1 = [x|1] @ [W1^T ; b1]  via V_WMMA_F32_16X16X4_F32 (K padded 9->12)
// Stage 2: per-thread 25-step LIF recurrence, h-outer/t-inner (mem1[h] independent),
//          spike contributions accumulated into 25x2 register accumulators,
//          using reset1_{t+1} == spk1_t (both Heaviside(mem1 - 1)).

#define BETA   0.95f
#define THR    1.0f
#define NSTEP  25
#define NIN    9
#define NHID   100
#define NOUT   2
#define BATCH  65536
#define ROWS   128          // batch rows per block (4 wave32)
#define KPAD   12           // 9 inputs + bias-row + 2 zero rows
#define NPAD   112          // 100 -> 7 tiles of 16
#define CSTRIDE 113         // odd stride: conflict-free LDS reads

typedef __attribute__((ext_vector_type(2))) float v2f;
typedef __attribute__((ext_vector_type(8))) float v8f;

__global__ __launch_bounds__(ROWS) void snn_lif_kernel(
    const float* __restrict__ x,  const float* __restrict__ W1,
    const float* __restrict__ b1, const float* __restrict__ W2,
    const float* __restrict__ b2, float* __restrict__ out)
{
    __shared__ float xt[ROWS * KPAD];      // [row][k], k=9 is 1.0 (bias), 10..11 zero
    __shared__ float w1t[KPAD * NPAD];     // [k][n] = W1[n][k]; row 9 = b1; rest 0
    __shared__ float cur1s[ROWS * CSTRIDE];
    __shared__ float w2s[2 * NHID];

    const int tid = threadIdx.x;
    const int rowBlk = blockIdx.x * ROWS;

    // ---------------- stage 0: cooperative loads into LDS ----------------
    {
        const float* xsrc = x + (size_t)rowBlk * NIN;
        #pragma unroll
        for (int i = 0; i < NIN; ++i) {           // 128*9 contiguous floats, coalesced
            int idx = tid + i * ROWS;
            xt[(idx / NIN) * KPAD + (idx % NIN)] = xsrc[idx];
        }
        xt[tid * KPAD +  9] = 1.0f;               // bias lane
        xt[tid * KPAD + 10] = 0.0f;
        xt[tid * KPAD + 11] = 0.0f;

        for (int idx = tid; idx < KPAD * NPAD; idx += ROWS) {
            int k = idx / NPAD, n = idx % NPAD;
            float v = 0.0f;
            if (n < NHID) {
                if (k < NIN)       v = W1[n * NIN + k];
                else if (k == NIN) v = b1[n];
            }
            w1t[idx] = v;
        }
        for (int idx = tid; idx < 2 * NHID; idx += ROWS)
            w2s[idx] = W2[idx];                   // W2 is [2][100] row-major
    }
    __syncthreads();

    // ---------------- stage 1: WMMA fp32 GEMM for cur1 ----------------
    {
        const int lane = tid & 31;
        const int wave = tid >> 5;
        const int l15  = lane & 15;
        const int kh   = (lane >> 4) * 2;         // lanes 0-15: K+{0,1}; 16-31: K+{2,3}

        #pragma unroll
        for (int rt = 0; rt < 2; ++rt) {
            const int rowBase = (wave * 2 + rt) * 16;
            #pragma unroll
            for (int ct = 0; ct < 7; ++ct) {
                const int colBase = ct * 16;
                v8f acc = {0.f, 0.f, 0.f, 0.f, 0.f, 0.f, 0.f, 0.f};
                #pragma unroll
                for (int ks = 0; ks < 3; ++ks) {
                    const int k0 = ks * 4 + kh;
                    v2f a = { xt[(rowBase + l15) * KPAD + k0],
                              xt[(rowBase + l15) * KPAD + k0 + 1] };
                    v2f b = { w1t[ k0      * NPAD + colBase + l15],
                              w1t[(k0 + 1) * NPAD + colBase + l15] };
                    acc = __builtin_amdgcn_wmma_f32_16x16x4_f32(
                              false, a, false, b, (short)0, acc, false, false);
                }
                const int srow = rowBase + ((lane >> 4) << 3);  // C/D layout: +8 for hi lanes
                #pragma unroll
                for (int i = 0; i < 8; ++i)
                    cur1s[(srow + i) * CSTRIDE + colBase + l15] = acc[i];
            }
        }
    }
    __syncthreads();

    // ---------------- stage 2: per-thread LIF recurrence ----------------
    float acc0[NSTEP], acc1[NSTEP];
    #pragma unroll
    for (int t = 0; t < NSTEP; ++t) { acc0[t] = 0.0f; acc1[t] = 0.0f; }

    const float* crow = &cur1s[tid * CSTRIDE];
    for (int h = 0; h < NHID; ++h) {
        const float ch = crow[h];
        const float wa = w2s[h];           // broadcast LDS read
        const float wb = w2s[NHID + h];
        float m = 0.0f, sp = 0.0f;         // sp doubles as reset for next step
        #pragma unroll
        for (int t = 0; t < NSTEP; ++t) {
            m  = __builtin_fmaf(BETA, m, ch - sp);   // beta*m + cur1 - reset*THR
            sp = (m > THR) ? 1.0f : 0.0f;            // spike (== next reset)
            acc0[t] = __builtin_fmaf(sp, wa, acc0[t]);
            acc1[t] = __builtin_fmaf(sp, wb, acc1[t]);
        }
    }

    const float b2a = b2[0], b2b = b2[1];
    float m2a = 0.0f, m2b = 0.0f;
    float2* o2 = (float2*)out;
    const size_t row = (size_t)rowBlk + tid;
    #pragma unroll
    for (int t = 0; t < NSTEP; ++t) {
        const float ra = (m2a > THR) ? 1.0f : 0.0f;  // reset from previous mem2
        const float rb = (m2b > THR) ? 1.0f : 0.0f;
        m2a = __builtin_fmaf(BETA, m2a, acc0[t] + b2a) - ra;
        m2b = __builtin_fmaf(BETA, m2b, acc1[t] + b2b) - rb;
        float2 v; v.x = m2a; v.y = m2b;
        o2[(size_t)t * BATCH + row] = v;             // out[t][row][0..1]
    }
}

extern "C" void kernel_launch(void* const* d_in, const int* in_sizes, int n_in,
                              void* d_out, int out_size, void* d_ws, size_t ws_size,
                              hipStream_t stream) {
    (void)in_sizes; (void)n_in; (void)d_ws; (void)ws_size; (void)out_size;
    const float* x  = (const float*)d_in[0];
    const float* W1 = (const float*)d_in[1];
    const float* b1 = (const float*)d_in[2];
    const float* W2 = (const float*)d_in[3];
    const float* b2 = (const float*)d_in[4];
    float* out = (float*)d_out;

    dim3 grid(BATCH / ROWS), block(ROWS);
    snn_lif_kernel<<<grid, block, 0, stream>>>(x, W1, b1, W2, b2, out);
}